// BevPoolV2_83021717832043
// MI455X (gfx1250) — compile-verified
//
#include <hip/hip_runtime.h>
#include <stdint.h>

// BEVPoolV2 on gfx1250: TDM gather-mode feat-row fetch into LDS (16 rows per
// descriptor, double buffered, TENSORcnt-tracked), index fetch pipelined one
// stage ahead through SMEM, per-lane float4 FMA accumulation, one wave per
// bev interval.

typedef unsigned int v4u __attribute__((ext_vector_type(4)));
typedef int          v8i __attribute__((ext_vector_type(8)));
typedef int          v4i __attribute__((ext_vector_type(4)));

static constexpr int CH       = 128;             // feature channels
static constexpr int SPATIAL  = 128 * 128;       // Z*HO*WO = 16384
static constexpr int NFEATROW = 2 * 6 * 32 * 88; // 33792 feat rows (< 2^16)
static constexpr int WAVES    = 8;               // waves per 256-thread block
static constexpr int CHUNK    = 16;              // rows per TDM gather descriptor

__device__ __forceinline__ int rfl(int x) { return __builtin_amdgcn_readfirstlane(x); }
__device__ __forceinline__ int imin(int a, int b) { return a < b ? a : b; }

// Fetch 16 row indices (scalar loads) and pack into 8 dwords of 16-bit pairs.
// FULL=true: unguarded consecutive reads (mergeable s_loads).
// FULL=false: addresses clamped to plast; slots past the chunk's n are unused
// downstream because tile_dim1 = n.
template <bool FULL>
__device__ __forceinline__ void load_pack_idx(const int* __restrict__ ranks_feat,
                                              int p, int plast, int* pk) {
  int idx[CHUNK];
#pragma unroll
  for (int j = 0; j < CHUNK; ++j)
    idx[j] = rfl(ranks_feat[FULL ? (p + j) : imin(p + j, plast)]);
#pragma unroll
  for (int k = 0; k < 8; ++k)
    pk[k] = (idx[2 * k + 1] << 16) | (idx[2 * k] & 0xFFFF);
}

// Issue one TDM gather: fetch n (<=16) rows of 128 f32 from `feat` into LDS at
// byte offset lds_off. D# layout per CDNA5 ISA ch8 (gather mode, 16-bit idx).
__device__ __forceinline__ void tdm_issue(uint32_t lds_off, uint64_t gbase,
                                          const int* pk, int n) {
  v4u g0;
  g0[0] = 0x80000001u;                         // gather_mode=1, 16b idx, count=1
  g0[1] = lds_off;                             // lds_addr (bytes)
  g0[2] = (uint32_t)gbase;                     // global_addr[31:0]
  g0[3] = (uint32_t)((gbase >> 32) & 0x01FFFFFFull) | (2u << 30); // [56:32] | type=2

  v8i g1;
  g1[0] = 0x00020000;                          // wg_mask=0, data_size=2 (4 bytes)
  g1[1] = (int)(128u << 16);                   // tensor_dim0 = 128 (bits 79:48)
  g1[2] = (int)(((uint32_t)(NFEATROW & 0xFFFF)) << 16); // tensor_dim1 lo16
  g1[3] = (int)(128u << 16) | (NFEATROW >> 16);         // tile_dim0=128 | td1 hi16
  g1[4] = n;                                   // tile_dim1 = #valid gather indices
  g1[5] = 128;                                 // tensor_dim0_stride = 128 elems
  g1[6] = 0;
  g1[7] = 0;

  v4i g2, g3;                                  // row_index_0..15 (16-bit pairs)
#pragma unroll
  for (int k = 0; k < 4; ++k) {
    g2[k] = pk[k];
    g3[k] = pk[4 + k];
  }

#if defined(__clang_major__) && (__clang_major__ >= 23)
  v8i gz = (v8i){0, 0, 0, 0, 0, 0, 0, 0};
  __builtin_amdgcn_tensor_load_to_lds(g0, g1, g2, g3, gz, 0);
#else
  __builtin_amdgcn_tensor_load_to_lds(g0, g1, g2, g3, 0);
#endif
}

__global__ __launch_bounds__(256) void bev_pool_v2_kernel(
    const float* __restrict__ depth,
    const float* __restrict__ feat,
    const int* __restrict__ ranks_depth,
    const int* __restrict__ ranks_feat,
    const int* __restrict__ ranks_bev,
    const int* __restrict__ interval_starts,
    const int* __restrict__ interval_lengths,
    float* __restrict__ out,
    int n_intervals,
    int n_points) {
  __shared__ float smem[WAVES][2][CHUNK * CH]; // 128 KB: 8 waves x 2 bufs x 8 KB
  const int wv   = (int)(threadIdx.x >> 5);
  const int lane = (int)(threadIdx.x & 31);
  const int gw   = (int)blockIdx.x * WAVES + wv;
  if (gw >= n_intervals) return;

  const int start = rfl(interval_starts[gw]);
  const int len   = rfl(interval_lengths[gw]);
  const int cell  = rfl(ranks_bev[start]);
  const int plast = n_points - 1;

  float4 acc = make_float4(0.f, 0.f, 0.f, 0.f);

  const uint64_t gbase = (uint64_t)(uintptr_t)feat;
  const uint32_t lds0 = (uint32_t)rfl((int)(uint32_t)(uintptr_t)(&smem[wv][0][0]));
  const uint32_t lds1 = (uint32_t)rfl((int)(uint32_t)(uintptr_t)(&smem[wv][1][0]));

  const int nchunks = (len + CHUNK - 1) / CHUNK;

  // Pipeline stage registers: packed indices for the next chunk to issue.
  int pk[8];

  // Prologue: fetch + issue chunk 0 into buffer 0.
  if (len >= CHUNK)
    load_pack_idx<true>(ranks_feat, start, plast, pk);
  else
    load_pack_idx<false>(ranks_feat, start, plast, pk);
  tdm_issue(lds0, gbase, pk, imin(CHUNK, len));

  // Prefetch indices for chunk 1 (overlaps with chunk 0's TDM).
  if (nchunks > 1) {
    if (len - CHUNK >= CHUNK)
      load_pack_idx<true>(ranks_feat, start + CHUNK, plast, pk);
    else
      load_pack_idx<false>(ranks_feat, start + CHUNK, plast, pk);
  }

  for (int c = 0; c < nchunks; ++c) {
    const int p = start + c * CHUNK;
    const int n = imin(CHUNK, len - c * CHUNK);
    if (c + 1 < nchunks) {
      // Issue chunk c+1 with already-fetched indices, prefetch indices for
      // chunk c+2 (their KMcnt wait overlaps with consuming chunk c), then
      // wait for chunk c's TDM.
      tdm_issue(((c + 1) & 1) ? lds1 : lds0, gbase, pk,
                imin(CHUNK, len - (c + 1) * CHUNK));
      if (c + 2 < nchunks) {
        const int p2 = start + (c + 2) * CHUNK;
        if (len - (c + 2) * CHUNK >= CHUNK)
          load_pack_idx<true>(ranks_feat, p2, plast, pk);
        else
          load_pack_idx<false>(ranks_feat, p2, plast, pk);
      }
      __builtin_amdgcn_s_wait_tensorcnt(1);
    } else {
      __builtin_amdgcn_s_wait_tensorcnt(0);
    }

    const float* buf = (c & 1) ? &smem[wv][1][0] : &smem[wv][0][0];
    if (n == CHUNK) {
      // Hot path: unguarded, consecutive ranks_depth reads (mergeable s_loads).
#pragma unroll
      for (int j = 0; j < CHUNK; ++j) {
        const float dj = depth[ranks_depth[p + j]];
        const float4 f = *(const float4*)(buf + j * CH + lane * 4); // ds_load_b128
        acc.x = fmaf(dj, f.x, acc.x);
        acc.y = fmaf(dj, f.y, acc.y);
        acc.z = fmaf(dj, f.z, acc.z);
        acc.w = fmaf(dj, f.w, acc.w);
      }
    } else {
      for (int j = 0; j < n; ++j) {
        const float dj = depth[ranks_depth[p + j]];
        const float4 f = *(const float4*)(buf + j * CH + lane * 4);
        acc.x = fmaf(dj, f.x, acc.x);
        acc.y = fmaf(dj, f.y, acc.y);
        acc.z = fmaf(dj, f.z, acc.z);
        acc.w = fmaf(dj, f.w, acc.w);
      }
    }
  }

  // Output layout [B, C, Z, HO, WO]: cell = b*16384 + spatial.
  const int b  = cell >> 14;
  const int sp = cell & (SPATIAL - 1);
  float* o = out + (size_t)b * CH * SPATIAL + sp;
  const int ch = lane * 4;
  o[(size_t)(ch + 0) * SPATIAL] = acc.x;
  o[(size_t)(ch + 1) * SPATIAL] = acc.y;
  o[(size_t)(ch + 2) * SPATIAL] = acc.z;
  o[(size_t)(ch + 3) * SPATIAL] = acc.w;
}

extern "C" void kernel_launch(void* const* d_in, const int* in_sizes, int n_in,
                              void* d_out, int out_size, void* d_ws, size_t ws_size,
                              hipStream_t stream) {
  const float* depth = (const float*)d_in[0];
  const float* feat  = (const float*)d_in[1];
  const int* ranks_depth      = (const int*)d_in[2];
  const int* ranks_feat       = (const int*)d_in[3];
  const int* ranks_bev        = (const int*)d_in[4];
  const int* interval_starts  = (const int*)d_in[5];
  const int* interval_lengths = (const int*)d_in[6];
  float* out = (float*)d_out;

  const int n_intervals = in_sizes[5];
  const int n_points    = in_sizes[2];

  // Empty BEV cells must be zero (harness poisons d_out).
  hipMemsetAsync(out, 0, (size_t)out_size * sizeof(float), stream);

  const int blocks = (n_intervals + WAVES - 1) / WAVES;
  bev_pool_v2_kernel<<<blocks, 256, 0, stream>>>(
      depth, feat, ranks_depth, ranks_feat, ranks_bev,
      interval_starts, interval_lengths, out, n_intervals, n_points);
}